// FlameShape_76098230550749
// MI455X (gfx1250) — compile-verified
//
#include <hip/hip_runtime.h>

typedef float v2f __attribute__((ext_vector_type(2)));
typedef float v8f __attribute__((ext_vector_type(8)));

#define NB 2048
#define NV 5023
#define NJ 5
#define KTOT 436              // 400 betas + 36 pose features
#define N3 15069              // NV*3
#define ROT_OFF ((size_t)NB * NV * 3)   // 30,861,312 floats

// workspace layout (in floats)
#define WS_JS 0                         // [401*16]: JS rows l=0..399 (16-padded, 15 used), Jt at row 400
#define WS_AF 6416                      // [NB * KTOT]
#define WS_AW (6416 + NB * KTOT)        // [NB * 60]  (5 joints x 3x4 affine)

// ---------------- zero scratch ----------------
__global__ void kZero(float* p, int n) {
    int t = blockIdx.x * 256 + threadIdx.x;
    if (t < n) p[t] = 0.f;
}

// ---------------- JS[l,jk] = sum_v jr[j,v]*shape_dirs[v,k,l]; Jt[jk] = sum_v jr[j,v]*vt[v,k] ----------------
__global__ void kA(const float* __restrict__ jr, const float* __restrict__ SD,
                   const float* __restrict__ VT, float* __restrict__ JS) {
    int jk = blockIdx.x;              // 0..14
    int chunk = blockIdx.y;           // 0..7
    int j = jk / 3, k = jk % 3;
    int tid = threadIdx.x;            // 256
    int v0 = chunk * 628;
    int v1 = v0 + 628; if (v1 > NV) v1 = NV;

    float acc0 = 0.f, acc1 = 0.f;
    for (int v = v0; v < v1; ++v) {
        float w = jr[j * NV + v];
        const float* sp = SD + ((size_t)(v * 3 + k)) * 400;
        acc0 += w * sp[tid];
        if (tid + 256 < 400) acc1 += w * sp[tid + 256];
    }
    atomicAdd(&JS[tid * 16 + jk], acc0);
    if (tid + 256 < 400) atomicAdd(&JS[(tid + 256) * 16 + jk], acc1);

    // Jt partial via LDS tree reduction
    float pa = 0.f;
    for (int v = v0 + tid; v < v1; v += 256) pa += jr[j * NV + v] * VT[v * 3 + k];
    __shared__ float red[256];
    red[tid] = pa;
    __syncthreads();
    for (int s = 128; s > 0; s >>= 1) {
        if (tid < s) red[tid] += red[tid + s];
        __syncthreads();
    }
    if (tid == 0) atomicAdd(&JS[400 * 16 + jk], red[0]);
}

// ---------------- pack betas = [shape | expr] into Af[:, 0:400] ----------------
__global__ void kPack(const float* __restrict__ shape, const float* __restrict__ expr,
                      float* __restrict__ Af) {
    int t = blockIdx.x * 256 + threadIdx.x;
    if (t >= NB * 400) return;
    int b = t / 400, i = t - b * 400;
    Af[(size_t)b * KTOT + i] = (i < 300) ? shape[b * 300 + i] : expr[b * 100 + (i - 300)];
}

__device__ __forceinline__ void compose(const float* Pr, const float* Pt,
                                        const float* Rm, const float* tt,
                                        float* Cr, float* Ct) {
#pragma unroll
    for (int r = 0; r < 3; ++r) {
#pragma unroll
        for (int c = 0; c < 3; ++c)
            Cr[r * 3 + c] = Pr[r * 3 + 0] * Rm[0 * 3 + c] + Pr[r * 3 + 1] * Rm[1 * 3 + c] +
                            Pr[r * 3 + 2] * Rm[2 * 3 + c];
        Ct[r] = Pr[r * 3 + 0] * tt[0] + Pr[r * 3 + 1] * tt[1] + Pr[r * 3 + 2] * tt[2] + Pt[r];
    }
}

// ---------------- per-batch: joints, Rodrigues, pose features, chain, relative transforms ----------------
__global__ void kB(float* __restrict__ Af, const float* __restrict__ gp,
                   const float* __restrict__ np_, const float* __restrict__ jp,
                   const float* __restrict__ ep, const float* __restrict__ JS,
                   float* __restrict__ Aw, float* __restrict__ rotOut) {
    int b = blockIdx.x * blockDim.x + threadIdx.x;
    if (b >= NB) return;

    // joints[jk] = Jt[jk] + sum_l betas[b,l] * JS[l,jk]
    float jnt[15];
#pragma unroll
    for (int i = 0; i < 15; ++i) jnt[i] = 0.f;
    const float* af = Af + (size_t)b * KTOT;
    for (int l = 0; l < 400; ++l) {
        float beta = af[l];
        const float* js = JS + l * 16;
#pragma unroll
        for (int i = 0; i < 15; ++i) jnt[i] += beta * js[i];
    }
#pragma unroll
    for (int i = 0; i < 15; ++i) jnt[i] += JS[400 * 16 + i];

    // full pose [15]
    float pose[15];
    pose[0] = gp[b * 3 + 0]; pose[1] = gp[b * 3 + 1]; pose[2] = gp[b * 3 + 2];
    pose[3] = np_[b * 3 + 0]; pose[4] = np_[b * 3 + 1]; pose[5] = np_[b * 3 + 2];
    pose[6] = jp[b * 3 + 0]; pose[7] = jp[b * 3 + 1]; pose[8] = jp[b * 3 + 2];
#pragma unroll
    for (int i = 0; i < 6; ++i) pose[9 + i] = ep[b * 6 + i];

    // Rodrigues (reference convention: angle from rvec+1e-8, axis = rvec/angle)
    float R[5][9];
#pragma unroll
    for (int j = 0; j < 5; ++j) {
        float x = pose[3 * j], y = pose[3 * j + 1], z = pose[3 * j + 2];
        float xa = x + 1e-8f, ya = y + 1e-8f, za = z + 1e-8f;
        float ang = sqrtf(xa * xa + ya * ya + za * za);
        float ax = x / ang, ay = y / ang, az = z / ang;
        float s = sinf(ang), c = cosf(ang), c1 = 1.f - c;
        R[j][0] = 1.f - c1 * (ay * ay + az * az);
        R[j][1] = -s * az + c1 * ax * ay;
        R[j][2] =  s * ay + c1 * ax * az;
        R[j][3] =  s * az + c1 * ax * ay;
        R[j][4] = 1.f - c1 * (ax * ax + az * az);
        R[j][5] = -s * ax + c1 * ay * az;
        R[j][6] = -s * ay + c1 * ax * az;
        R[j][7] =  s * ax + c1 * ay * az;
        R[j][8] = 1.f - c1 * (ax * ax + ay * ay);
    }

    // rot_mats output [B,J,3,3]
    float* ro = rotOut + (size_t)b * 45;
#pragma unroll
    for (int j = 0; j < 5; ++j)
#pragma unroll
        for (int i = 0; i < 9; ++i) ro[j * 9 + i] = R[j][i];

    // pose features -> Af[:, 400:436]
    float* pf = Af + (size_t)b * KTOT + 400;
#pragma unroll
    for (int j = 1; j < 5; ++j)
#pragma unroll
        for (int i = 0; i < 9; ++i)
            pf[(j - 1) * 9 + i] = R[j][i] - ((i == 0 || i == 4 || i == 8) ? 1.f : 0.f);

    // relative joints (parents = {-1,0,1,1,1}, matching reference PARENTS)
    float rel[5][3];
#pragma unroll
    for (int k = 0; k < 3; ++k) {
        rel[0][k] = jnt[k];
        rel[1][k] = jnt[3 + k]  - jnt[k];
        rel[2][k] = jnt[6 + k]  - jnt[3 + k];
        rel[3][k] = jnt[9 + k]  - jnt[3 + k];
        rel[4][k] = jnt[12 + k] - jnt[3 + k];
    }

    // kinematic chain (affine 3x4 compose; bottom row implicit [0,0,0,1])
    float cr[5][9], ctl[5][3];
#pragma unroll
    for (int i = 0; i < 9; ++i) cr[0][i] = R[0][i];
#pragma unroll
    for (int k = 0; k < 3; ++k) ctl[0][k] = rel[0][k];
    compose(cr[0], ctl[0], R[1], rel[1], cr[1], ctl[1]);
    compose(cr[1], ctl[1], R[2], rel[2], cr[2], ctl[2]);
    compose(cr[1], ctl[1], R[3], rel[3], cr[3], ctl[3]);
    compose(cr[1], ctl[1], R[4], rel[4], cr[4], ctl[4]);

    // relative transforms A (trans -= rot * rest_joint); store 3x4 row-major per joint
    float* aw = Aw + (size_t)b * 60;
#pragma unroll
    for (int j = 0; j < 5; ++j) {
        float jx = jnt[3 * j], jy = jnt[3 * j + 1], jz = jnt[3 * j + 2];
#pragma unroll
        for (int r = 0; r < 3; ++r) {
            aw[j * 12 + r * 4 + 0] = cr[j][r * 3 + 0];
            aw[j * 12 + r * 4 + 1] = cr[j][r * 3 + 1];
            aw[j * 12 + r * 4 + 2] = cr[j][r * 3 + 2];
            aw[j * 12 + r * 4 + 3] = ctl[j][r] -
                (cr[j][r * 3 + 0] * jx + cr[j][r * 3 + 1] * jy + cr[j][r * 3 + 2] * jz);
        }
    }
}

// ---------------- fused WMMA GEMM (v_posed) + LBS epilogue ----------------
// grid: (79, 128); block: 128 (4 waves). blockIdx.y = M-tile (16 batches),
// wave -> N-tile (16 vertices). 3 accumulators = x/y/z coordinate planes.
__global__ void __launch_bounds__(128) kC(const float* __restrict__ Af,
                                          const float* __restrict__ SD,
                                          const float* __restrict__ PD,
                                          const float* __restrict__ VT,
                                          const float* __restrict__ W,
                                          const float* __restrict__ Aw,
                                          float* __restrict__ out) {
    __shared__ float sAw[16 * 60];
    int tid = threadIdx.x;
    int b0 = blockIdx.y * 16;
    for (int i = tid; i < 960; i += 128) sAw[i] = Aw[(size_t)b0 * 60 + i];
    __syncthreads();

    int wave = tid >> 5, lane = tid & 31;
    int nt = blockIdx.x * 4 + wave;
    if (nt >= 314) return;                 // wave-uniform: EXEC stays all-1s for active waves
    int row = lane & 15, half = lane >> 4;
    int v0 = nt * 16;
    int vr = v0 + row;
    int v = (vr < NV) ? vr : (NV - 1);     // clamp for tail tile

    // A fragment source: Af[(b0+row), k0 + 2*half + {0,1}]
    const float* ap = Af + (size_t)(b0 + row) * KTOT + 2 * half;
    // B fragment sources (shape part): SD[(v*3+c)*400 + k0 + 2*half + {0,1}]
    const float* sp0 = SD + (size_t)(v * 3 + 0) * 400 + 2 * half;
    const float* sp1 = SD + (size_t)(v * 3 + 1) * 400 + 2 * half;
    const float* sp2 = SD + (size_t)(v * 3 + 2) * 400 + 2 * half;

    v8f a0 = {}, a1 = {}, a2 = {};

#pragma unroll 2
    for (int s = 0; s < 100; ++s) {        // K = 0..399 (shape/expr blendshapes)
        v2f af  = *(const v2f*)(ap  + 4 * s);
        v2f bf0 = *(const v2f*)(sp0 + 4 * s);
        v2f bf1 = *(const v2f*)(sp1 + 4 * s);
        v2f bf2 = *(const v2f*)(sp2 + 4 * s);
        a0 = __builtin_amdgcn_wmma_f32_16x16x4_f32(false, af, false, bf0, (short)0, a0, false, false);
        a1 = __builtin_amdgcn_wmma_f32_16x16x4_f32(false, af, false, bf1, (short)0, a1, false, false);
        a2 = __builtin_amdgcn_wmma_f32_16x16x4_f32(false, af, false, bf2, (short)0, a2, false, false);
    }

    // K = 400..435 (pose blendshapes); pose_dirs is [36, V*3] row-major
    const float* pp = PD + (size_t)(2 * half) * N3 + (size_t)v * 3;
#pragma unroll
    for (int s = 0; s < 9; ++s) {
        v2f af = *(const v2f*)(ap + 400 + 4 * s);
        const float* p = pp + (size_t)(4 * s) * N3;
        v2f bf0, bf1, bf2;
        bf0[0] = p[0]; bf0[1] = p[N3 + 0];
        bf1[0] = p[1]; bf1[1] = p[N3 + 1];
        bf2[0] = p[2]; bf2[1] = p[N3 + 2];
        a0 = __builtin_amdgcn_wmma_f32_16x16x4_f32(false, af, false, bf0, (short)0, a0, false, false);
        a1 = __builtin_amdgcn_wmma_f32_16x16x4_f32(false, af, false, bf1, (short)0, a1, false, false);
        a2 = __builtin_amdgcn_wmma_f32_16x16x4_f32(false, af, false, bf2, (short)0, a2, false, false);
    }

    // LBS epilogue: all three coords of vertex v live in this lane (C/D: lane=N, VGPR=M)
    float vtx = VT[v * 3 + 0], vty = VT[v * 3 + 1], vtz = VT[v * 3 + 2];
    float wj[5];
#pragma unroll
    for (int j = 0; j < 5; ++j) wj[j] = W[(size_t)v * 5 + j];
    bool valid = (vr < NV);

#pragma unroll
    for (int r = 0; r < 8; ++r) {
        int bl = r + 8 * half;             // local batch row (C/D layout)
        float px = a0[r] + vtx, py = a1[r] + vty, pz = a2[r] + vtz;
        float ox = 0.f, oy = 0.f, oz = 0.f;
#pragma unroll
        for (int j = 0; j < 5; ++j) {
            const float* t = &sAw[bl * 60 + j * 12];
            float wv = wj[j];
            ox += wv * (t[0] * px + t[1] * py + t[2]  * pz + t[3]);
            oy += wv * (t[4] * px + t[5] * py + t[6]  * pz + t[7]);
            oz += wv * (t[8] * px + t[9] * py + t[10] * pz + t[11]);
        }
        if (valid) {
            size_t o = ((size_t)(b0 + bl) * NV + vr) * 3;
            out[o + 0] = ox; out[o + 1] = oy; out[o + 2] = oz;
        }
    }
}

extern "C" void kernel_launch(void* const* d_in, const int* in_sizes, int n_in,
                              void* d_out, int out_size, void* d_ws, size_t ws_size,
                              hipStream_t stream) {
    const float* shape = (const float*)d_in[0];
    const float* expr  = (const float*)d_in[1];
    const float* gp    = (const float*)d_in[2];
    const float* np_   = (const float*)d_in[3];
    const float* jp    = (const float*)d_in[4];
    const float* ep    = (const float*)d_in[5];
    const float* VT    = (const float*)d_in[6];
    const float* SD    = (const float*)d_in[7];
    const float* PD    = (const float*)d_in[8];
    const float* JR    = (const float*)d_in[9];
    const float* W     = (const float*)d_in[10];
    // d_in[11] = parents; constant FLAME kintree {-1,0,1,1,1}, hardcoded like the reference

    float* out = (float*)d_out;
    float* ws  = (float*)d_ws;
    float* JS  = ws + WS_JS;
    float* Af  = ws + WS_AF;
    float* Aw  = ws + WS_AW;

    kZero<<<(6416 + 255) / 256, 256, 0, stream>>>(JS, 6416);
    kA<<<dim3(15, 8), 256, 0, stream>>>(JR, SD, VT, JS);
    kPack<<<(NB * 400 + 255) / 256, 256, 0, stream>>>(shape, expr, Af);
    kB<<<NB / 256, 256, 0, stream>>>(Af, gp, np_, jp, ep, JS, Aw, out + ROT_OFF);
    kC<<<dim3(79, 128), 128, 0, stream>>>(Af, SD, PD, VT, W, Aw, out);
}